// CustomPositionLoss_35115652612313
// MI455X (gfx1250) — compile-verified
//
#include <hip/hip_runtime.h>

#define DT 0.005f
#define TPB 256
#define NWAVES (TPB / 32)
#define NBLOCKS 1024

typedef __attribute__((ext_vector_type(2))) float v2f;
typedef __attribute__((ext_vector_type(8))) float v8f;

// Sum a value across all 32 lanes of a wave using V_WMMA_F32_16X16X4_F32.
// A (16x4): lane L<16 -> A[L][0]=x, A[L][1]=0 ; lane L>=16 -> A[L-16][2]=x, A[L-16][3]=0
// B (4x16): all ones.  D[m][n] = sum_k A[m][k]  (row sums, replicated over n).
// Lane L<16 holds D[0..7][n=L] in its 8 result VGPRs; lane L>=16 holds D[8..15][n=L-16].
// Sum of the 8 result VGPRs gives the half-wave sum; one SWAPX16 ds_swizzle
// merges the halves. Exact f32, fixed order -> deterministic.
// Requires EXEC == all ones (guaranteed: called uniformly from full blocks).
__device__ __forceinline__ float wave_sum32_wmma(float x) {
  v2f a; a[0] = x;    a[1] = 0.0f;
  v2f b; b[0] = 1.0f; b[1] = 1.0f;
  v8f c = {};
  c = __builtin_amdgcn_wmma_f32_16x16x4_f32(
      /*neg_a=*/false, a, /*neg_b=*/false, b,
      /*c_mod=*/(short)0, c, /*reuse_a=*/false, /*reuse_b=*/false);
  float s = ((c[0] + c[1]) + (c[2] + c[3])) + ((c[4] + c[5]) + (c[6] + c[7]));
  // SWAPX16: group-of-32 swizzle, xor_mask=0x10, or_mask=0, and_mask=0x1f
  s += __int_as_float(__builtin_amdgcn_ds_swizzle(__float_as_int(s), 0x401F));
  return s;  // valid on every lane
}

// Block reduction: per-wave WMMA sum -> 8 LDS slots -> thread 0 straight-line sum.
__device__ __forceinline__ void block_reduce_store(float v, float scale,
                                                   float* __restrict__ dst) {
  const float wsum = wave_sum32_wmma(v);
  __shared__ float lds[NWAVES];
  const int lane = threadIdx.x & 31;
  const int wid  = threadIdx.x >> 5;
  if (lane == 0) lds[wid] = wsum;
  __syncthreads();
  if (threadIdx.x == 0) {
    const float s = ((lds[0] + lds[1]) + (lds[2] + lds[3])) +
                    ((lds[4] + lds[5]) + (lds[6] + lds[7]));
    *dst = s * scale;
  }
}

__global__ void __launch_bounds__(TPB)
pos_loss_partial_kernel(const float* __restrict__ quats,   // (B,4)
                        const float* __restrict__ tpos,    // (B,3)
                        const float* __restrict__ bias,    // (B,3)
                        const float* __restrict__ X,       // (B,S,F)
                        const float* __restrict__ pos_all, // (N,3)
                        const float* __restrict__ vel_all, // (N,3)
                        const float* __restrict__ grav,    // (3,)
                        const int*   __restrict__ idx,     // (B,)
                        const int*   __restrict__ seqp,    // scalar S
                        int B, int SF,                     // SF = S*F
                        float* __restrict__ partials)      // (gridDim.x,)
{
  const int tid      = blockIdx.x * TPB + threadIdx.x;
  const int nthreads = gridDim.x * TPB;

  const int   S    = seqp[0];
  const int   F    = SF / S;                 // feature count (6)
  const int   xoff = SF - F;                 // offset of last timestep's row
  const float hdt2 = 0.5f * DT * DT;
  const float gp0  = grav[0] * hdt2;
  const float gp1  = grav[1] * hdt2;
  const float gp2  = grav[2] * hdt2;

  float acc_sum = 0.0f;

  for (int bI = tid; bI < B; bI += nthreads) {
    const size_t b = (size_t)bI;

    const float4 qq = ((const float4*)quats)[b];
    const float q0 = qq.x, q1 = qq.y, q2 = qq.z, q3 = qq.w;

    // acc = batch_X[b, S-1, 0:3] - predicted_biases[b]
    const float* xb = X + b * (size_t)SF + xoff;
    const float a1 = xb[0] - bias[b * 3 + 0];
    const float a2 = xb[1] - bias[b * 3 + 1];
    const float a3 = xb[2] - bias[b * 3 + 2];

    // t = q (x) (0, a) ; u = t (x) conj(q)   [exact expansion of reference]
    const float t0 = -q1 * a1 - q2 * a2 - q3 * a3;
    const float t1 =  q0 * a1 + q2 * a3 - q3 * a2;
    const float t2 =  q0 * a2 - q1 * a3 + q3 * a1;
    const float t3 =  q0 * a3 + q1 * a2 - q2 * a1;
    const float u1 = -t0 * q1 + t1 * q0 - t2 * q3 + t3 * q2;
    const float u2 = -t0 * q2 + t1 * q3 + t2 * q0 - t3 * q1;
    const float u3 = -t0 * q3 - t1 * q2 + t2 * q1 + t3 * q0;

    // gather p0, v0 at clamp(idx - (S-1), 0)
    int i0 = idx[b] - (S - 1);
    if (i0 < 0) i0 = 0;
    const float* p0 = pos_all + 3 * (size_t)i0;
    const float* v0 = vel_all + 3 * (size_t)i0;

    const float pr1 = p0[0] + v0[0] * DT + gp0 + hdt2 * u1;
    const float pr2 = p0[1] + v0[1] * DT + gp1 + hdt2 * u2;
    const float pr3 = p0[2] + v0[2] * DT + gp2 + hdt2 * u3;

    const float d1 = pr1 - tpos[b * 3 + 0];
    const float d2 = pr2 - tpos[b * 3 + 1];
    const float d3 = pr3 - tpos[b * 3 + 2];

    const float ad1 = fabsf(d1), ad2 = fabsf(d2), ad3 = fabsf(d3);
    const float h1 = (ad1 < 1.0f) ? 0.5f * d1 * d1 : ad1 - 0.5f;
    const float h2 = (ad2 < 1.0f) ? 0.5f * d2 * d2 : ad2 - 0.5f;
    const float h3 = (ad3 < 1.0f) ? 0.5f * d3 * d3 : ad3 - 0.5f;

    acc_sum += (h1 + h2) + h3;
  }

  block_reduce_store(acc_sum, 1.0f, &partials[blockIdx.x]);
}

__global__ void __launch_bounds__(TPB)
pos_loss_finalize_kernel(const float* __restrict__ partials, int n,
                         float scale, float* __restrict__ out)
{
  float s = 0.0f;
  for (int i = threadIdx.x; i < n; i += TPB) s += partials[i];
  block_reduce_store(s, scale, out);
}

extern "C" void kernel_launch(void* const* d_in, const int* in_sizes, int n_in,
                              void* d_out, int out_size, void* d_ws, size_t ws_size,
                              hipStream_t stream) {
  const float* quats   = (const float*)d_in[0];  // (B,4)
  const float* tpos    = (const float*)d_in[1];  // (B,3)
  const float* bias    = (const float*)d_in[2];  // (B,3)
  const float* X       = (const float*)d_in[3];  // (B,S,F)
  const float* pos_all = (const float*)d_in[4];  // (N,3)
  const float* vel_all = (const float*)d_in[5];  // (N,3)
  const float* grav    = (const float*)d_in[6];  // (3,)
  const int*   idx     = (const int*)d_in[7];    // (B,)
  const int*   seqp    = (const int*)d_in[8];    // scalar

  const int B  = in_sizes[0] / 4;
  const int SF = (B > 0) ? (in_sizes[3] / B) : 0;  // S*F

  float* partials = (float*)d_ws;  // needs NBLOCKS*4 = 4KB of scratch

  pos_loss_partial_kernel<<<NBLOCKS, TPB, 0, stream>>>(
      quats, tpos, bias, X, pos_all, vel_all, grav, idx, seqp, B, SF, partials);

  const float scale = 1.0f / (3.0f * (float)B);
  pos_loss_finalize_kernel<<<1, TPB, 0, stream>>>(partials, NBLOCKS, scale,
                                                  (float*)d_out);
}